// PRNNLayer_22625887715870
// MI455X (gfx1250) — compile-verified
//
#include <hip/hip_runtime.h>

// ---------------------------------------------------------------------------
// PRNN (EXP-HYDRO) batched scan for MI455X / gfx1250.
//   B=8192 independent sequences, T=1024 steps, 3 inputs/step, 2 outputs/step.
//   grid = B/32 single-wave (wave32) workgroups; each lane owns one sequence.
//   Inputs staged global->LDS with double-buffered async-to-LDS b128 DMA
//   (ASYNCcnt / s_wait_asynccnt); outputs staged in LDS, stored coalesced.
//   t-loop unrolled x4 so state-independent math of future steps overlaps the
//   serial (s0,s1) transcendental chain (1 wave/SIMD -> ILP is the only cover).
// ---------------------------------------------------------------------------

#define T_LEN  1024
#define TC     32                      // timesteps per LDS chunk
#define NCH    (T_LEN / TC)            // 32 chunks
#define RS_IN  (TC * 3 + 4)            // padded input row stride (floats): 100
#define RS_OUT (TC * 2 + 4)            // padded output row stride (floats): 68
#define N4_IN  (TC * 3 / 4)            // float4 per input row: 24
#define LOADS_PER_CHUNK N4_IN          // 24 b128 async issues per lane/chunk
#define N4_OUT (TC * 2 / 4)            // float4 per output row: 16
#define CLIPV  100000.0f

#if defined(__HIP_DEVICE_COMPILE__) && defined(__gfx1250__)
#  if __has_builtin(__builtin_amdgcn_global_load_async_to_lds_b128)
#    define HAVE_ASYNC_LDS 1
#  endif
#endif

#if defined(HAVE_ASYNC_LDS)
typedef int v4i __attribute__((__vector_size__(16)));
typedef v4i __attribute__((address_space(1))) g_v4i;   // global (AS1) v4i
typedef v4i __attribute__((address_space(3))) l_v4i;   // LDS (AS3) v4i

__device__ __forceinline__ g_v4i* to_global(const void* p) {
  return (g_v4i*)(unsigned long long)p;                // flat==global numerically
}
__device__ __forceinline__ l_v4i* to_local(const void* p) {
  return (l_v4i*)(unsigned int)(unsigned long long)p;  // low 32b = LDS offset
}
#  if __has_builtin(__builtin_amdgcn_s_wait_asynccnt)
#    define WAIT_ASYNC(n) __builtin_amdgcn_s_wait_asynccnt(n)
#  else
#    define WAIT_ASYNC(n) asm volatile("s_wait_asynccnt %0" :: "i"(n) : "memory")
#  endif
#endif

// heaviside(x) = (tanh(5x)+1)/2 == 1/(1+exp(-10x))  -> v_exp_f32 + v_rcp_f32
__device__ __forceinline__ float heav(float x) {
  return __builtin_amdgcn_rcpf(1.0f + __expf(-10.0f * x));
}

__device__ __forceinline__ void soil(float s1, float pet,
                                     float f_, float smax_, float inv_smax,
                                     float qmax_, float k_, float omk,
                                     float& et, float& qsurf, float& qst, float& qgr) {
  const float hp = heav(s1);
  const float d  = s1 - smax_;                 // s1 - smax
  const float po = hp * heav(d);               // over
  const float pu = hp * heav(-d);              // under
  et             = po * pet + pu * pet * (s1 * inv_smax);
  const float qs = po * qmax_ + pu * qmax_ * __expf(-f_ * (smax_ - s1));
  qsurf          = po * d;
  qst            = qs * k_;
  qgr            = qs * omk;
}

// Stage one (32 rows x TC steps x 3) tile into LDS, coalesced.
__device__ __forceinline__ void load_chunk(const float* __restrict__ in,
                                           float* tile, int b0, int chunk,
                                           int lane, int B) {
  const int t0 = chunk * TC;
#pragma unroll
  for (int i = 0; i < LOADS_PER_CHUNK; ++i) {
    const int linear = i * 32 + lane;
    const int row    = linear / N4_IN;
    const int col    = linear - row * N4_IN;
    int gb = b0 + row; gb = (gb < B) ? gb : (B - 1);
    const float* src = in + ((size_t)gb * (T_LEN * 3) + (size_t)(t0 * 3 + col * 4));
    float*       dst = tile + row * RS_IN + col * 4;
#if defined(HAVE_ASYNC_LDS)
    __builtin_amdgcn_global_load_async_to_lds_b128(to_global(src), to_local(dst), 0, 0);
#else
    *(float4*)dst = *(const float4*)src;
#endif
  }
}

// Flush one (32 rows x TC steps x 2) LDS tile to global, coalesced 256B rows.
__device__ __forceinline__ void store_chunk(float* __restrict__ out,
                                            const float* tile, int b0, int chunk,
                                            int lane, int B) {
  const int t0 = chunk * TC;
#pragma unroll
  for (int i = 0; i < N4_OUT; ++i) {          // 16 iters: 512 float4 total
    const int linear = i * 32 + lane;
    const int row    = linear / N4_OUT;
    const int col    = linear - row * N4_OUT;
    const int gb     = b0 + row;
    const float4 v   = *(const float4*)(tile + row * RS_OUT + col * 4);
    if (gb < B)
      *(float4*)(out + ((size_t)gb * T_LEN + t0) * 2 + col * 4) = v;
  }
}

__global__ void __launch_bounds__(32) prnn_scan_kernel(
    const float* __restrict__ in,
    const float* __restrict__ p_f,    const float* __restrict__ p_smax,
    const float* __restrict__ p_qmax, const float* __restrict__ p_ddf,
    const float* __restrict__ p_tmin, const float* __restrict__ p_tmax,
    const float* __restrict__ p_k,
    float* __restrict__ out, int B)
{
  __shared__ float tin[2][32 * RS_IN];   // double-buffered input tiles (25.0 KB)
  __shared__ float tout[32 * RS_OUT];    // output tile (8.5 KB)

  const int lane = threadIdx.x;
  const int b0   = blockIdx.x * 32;

  const float f_    = p_f[0]    * 0.1f;
  const float smax_ = p_smax[0] * 1500.0f;
  const float qmax_ = p_qmax[0] * 50.0f;
  const float ddf_  = p_ddf[0]  * 5.0f;
  const float tmin_ = p_tmin[0] * -3.0f;
  const float tmax_ = p_tmax[0] * 3.0f;
  const float k_    = p_k[0]    * 0.1f;
  const float inv_smax = 1.0f / smax_;
  const float omk      = 1.0f - k_;

  float s0 = 0.0f, s1 = 0.0f;

  load_chunk(in, tin[0], b0, 0, lane, B);

  for (int c = 0; c < NCH; ++c) {
    if (c + 1 < NCH) {
      load_chunk(in, tin[(c + 1) & 1], b0, c + 1, lane, B);
#if defined(HAVE_ASYNC_LDS)
      WAIT_ASYNC(LOADS_PER_CHUNK);   // chunk c complete; chunk c+1 still in flight
#endif
    } else {
#if defined(HAVE_ASYNC_LDS)
      WAIT_ASYNC(0);
#endif
    }
    asm volatile("" ::: "memory");

    const float* __restrict__ row  = &tin[c & 1][lane * RS_IN];
    float*       __restrict__ orow = &tout[lane * RS_OUT];

#pragma unroll 4
    for (int t = 0; t < TC; ++t) {
      const float prcp = row[3 * t + 0];
      const float tm   = row[3 * t + 1];
      const float dayl = row[3 * t + 2];

      const float pet = 29.8f * (dayl * 24.0f) * 0.611f
                        * __expf(17.3f * tm / (tm + 237.3f)) / (tm + 273.2f);

      const float psnow = heav(tmin_ - tm) * prcp;
      const float prain = heav(tm - tmin_) * prcp;

      const float dtm = tm - tmax_;
      const float m   = heav(dtm) * heav(s0) * fminf(s0, ddf_ * dtm);

      float et, qsurf, qst, qgr;
      soil(s1, pet, f_, smax_, inv_smax, qmax_, k_, omk, et, qsurf, qst, qgr);

      const float ds0 = psnow - m;
      const float ds1 = prain + m - et - qsurf - qst - qgr;
      s0 += fminf(fmaxf(ds0, -CLIPV), CLIPV);
      s1 += fminf(fmaxf(ds1, -CLIPV), CLIPV);

      // outputs use the post-update state s1 (== s1_seq[t]) with this step's pet
      float et2, qsurf2, qst2, qgr2;
      soil(s1, pet, f_, smax_, inv_smax, qmax_, k_, omk, et2, qsurf2, qst2, qgr2);
      orow[2 * t + 0] = qsurf2 + qst2;
      orow[2 * t + 1] = qgr2;
    }

    store_chunk(out, tout, b0, c, lane, B);
  }
}

extern "C" void kernel_launch(void* const* d_in, const int* in_sizes, int n_in,
                              void* d_out, int out_size, void* d_ws, size_t ws_size,
                              hipStream_t stream) {
  const float* in     = (const float*)d_in[0];
  const float* p_f    = (const float*)d_in[1];
  const float* p_smax = (const float*)d_in[2];
  const float* p_qmax = (const float*)d_in[3];
  const float* p_ddf  = (const float*)d_in[4];
  const float* p_tmin = (const float*)d_in[5];
  const float* p_tmax = (const float*)d_in[6];
  const float* p_k    = (const float*)d_in[7];
  float* out = (float*)d_out;

  const int B    = in_sizes[0] / (T_LEN * 3);
  const int grid = (B + 31) / 32;
  prnn_scan_kernel<<<grid, 32, 0, stream>>>(in, p_f, p_smax, p_qmax, p_ddf,
                                            p_tmin, p_tmax, p_k, out, B);
  (void)n_in; (void)out_size; (void)d_ws; (void)ws_size;
}